// RBFInterpolator_43705587204608
// MI455X (gfx1250) — compile-verified
//
#include <hip/hip_runtime.h>

// Problem constants (from the reference)
#define N_X   8192
#define N_XP  4096
#define X_DIM 64

#define I_TILES (N_X / 16)   // 512
#define J_TILES (N_XP / 16)  // 256
#define SEGS    8            // i-range split for the density pass

typedef __attribute__((ext_vector_type(2))) float v2f;
typedef __attribute__((ext_vector_type(8))) float v8f;

// -------------------------------------------------------------------------
// Kernel 0: row squared norms + 1/scale^2
// ws layout (floats): x2[0:8192) | xp2[8192:12288) | part[12288:45056)
//                     | dens[45056:49152) | invd[49152:53248) | inv_s2[53248]
// -------------------------------------------------------------------------
__global__ __launch_bounds__(256) void k_pre(const float* __restrict__ x,
                                             const float* __restrict__ xp,
                                             const float* __restrict__ sls,
                                             float* __restrict__ x2,
                                             float* __restrict__ xp2,
                                             float* __restrict__ inv_s2) {
  int i = blockIdx.x * blockDim.x + threadIdx.x;
  if (i < N_X) {
    const float4* p = (const float4*)(x + (size_t)i * X_DIM);
    float s = 0.f;
#pragma unroll
    for (int k = 0; k < X_DIM / 4; ++k) {
      float4 v = p[k];
      s += v.x * v.x + v.y * v.y + v.z * v.z + v.w * v.w;
    }
    x2[i] = s;
  }
  if (i < N_XP) {
    const float4* p = (const float4*)(xp + (size_t)i * X_DIM);
    float s = 0.f;
#pragma unroll
    for (int k = 0; k < X_DIM / 4; ++k) {
      float4 v = p[k];
      s += v.x * v.x + v.y * v.y + v.z * v.z + v.w * v.w;
    }
    xp2[i] = s;
  }
  if (i == 0) {
    float v = *sls;
    float sp = (v > 20.0f) ? v : log1pf(__expf(v));  // stable softplus
    float scale = 1e-5f + sp;
    *inv_s2 = 1.0f / (scale * scale);
  }
}

// -------------------------------------------------------------------------
// Kernel 1: density partials. blockIdx.x = j-tile (16 rows of xp),
// blockIdx.y = i-segment. 8 waves/block, each wave owns a 16x16 WMMA tile
// and walks 8 i-tiles of its segment. Deterministic reductions only.
// -------------------------------------------------------------------------
__global__ __launch_bounds__(256) void k_density(const float* __restrict__ x,
                                                 const float* __restrict__ xp,
                                                 const float* __restrict__ x2,
                                                 const float* __restrict__ xp2,
                                                 const float* __restrict__ inv_s2p,
                                                 float* __restrict__ part) {
  __shared__ float s_red[8][16];

  const int j0   = blockIdx.x * 16;
  const int seg  = blockIdx.y;
  const int wave = threadIdx.x >> 5;
  const int lane = threadIdx.x & 31;
  const int hf   = lane >> 4;   // K-half selector for 16x16x4 f32 layout
  const int rw   = lane & 15;   // M for A fragment, N for B fragment

  const float inv_s2 = *inv_s2p;

  // A fragment: 16 chained K=4 steps, per lane elements (4k+2h, 4k+2h+1) of row j0+rw
  v2f af[16];
  {
    const float* ap = xp + (size_t)(j0 + rw) * X_DIM + 2 * hf;
#pragma unroll
    for (int kk = 0; kk < 16; ++kk) af[kk] = *(const v2f*)(ap + 4 * kk);
  }
  float xp2v[8];
#pragma unroll
  for (int r = 0; r < 8; ++r) xp2v[r] = xp2[j0 + r + 8 * hf];

  float esum[8];
#pragma unroll
  for (int r = 0; r < 8; ++r) esum[r] = 0.f;

  for (int t = 0; t < (I_TILES / SEGS) / 8; ++t) {       // 8 tiles per wave
    const int it = seg * (I_TILES / SEGS) + t * 8 + wave;
    const int i0 = it * 16;
    const float* bp = x + (size_t)(i0 + rw) * X_DIM + 2 * hf;
    v8f c = {};
#pragma unroll
    for (int kk = 0; kk < 16; ++kk) {
      v2f b = *(const v2f*)(bp + 4 * kk);
      c = __builtin_amdgcn_wmma_f32_16x16x4_f32(false, af[kk], false, b,
                                                (short)0, c, false, false);
    }
    const float x2v = x2[i0 + rw];
#pragma unroll
    for (int r = 0; r < 8; ++r) {
      float d2 = fmaxf(xp2v[r] + x2v - 2.0f * c[r], 0.0f);
      esum[r] += __expf(-d2 * inv_s2);
    }
  }

  // reduce over the 16 lanes of each half (xor masks 1,2,4,8 stay in-half)
#pragma unroll
  for (int r = 0; r < 8; ++r) {
    float v = esum[r];
    v += __shfl_xor(v, 1, 32);
    v += __shfl_xor(v, 2, 32);
    v += __shfl_xor(v, 4, 32);
    v += __shfl_xor(v, 8, 32);
    esum[r] = v;
  }
  if (rw == 0) {  // lane 0 -> rows 0..7, lane 16 -> rows 8..15
#pragma unroll
    for (int r = 0; r < 8; ++r) s_red[wave][r + 8 * hf] = esum[r];
  }
  __syncthreads();
  if (threadIdx.x < 16) {
    float s = 0.f;
#pragma unroll
    for (int w = 0; w < 8; ++w) s += s_red[w][threadIdx.x];  // fixed order
    part[(size_t)seg * N_XP + j0 + threadIdx.x] = s;
  }
}

// -------------------------------------------------------------------------
// Kernel 2: fold segment partials -> density and reciprocal (fixed order)
// -------------------------------------------------------------------------
__global__ __launch_bounds__(256) void k_reduce(const float* __restrict__ part,
                                                float* __restrict__ dens,
                                                float* __restrict__ invd) {
  int j = blockIdx.x * blockDim.x + threadIdx.x;
  if (j < N_XP) {
    float s = 0.f;
#pragma unroll
    for (int g = 0; g < SEGS; ++g) s += part[(size_t)g * N_XP + j];
    dens[j] = s;
    invd[j] = 1.0f / s;
  }
}

// -------------------------------------------------------------------------
// Kernel 3: recompute tiles, scale by 1/density, write weights (+ density)
// One wave per 16x16 tile; 8 tiles (same j-tile) per block.
// -------------------------------------------------------------------------
__global__ __launch_bounds__(256) void k_weights(const float* __restrict__ x,
                                                 const float* __restrict__ xp,
                                                 const float* __restrict__ x2,
                                                 const float* __restrict__ xp2,
                                                 const float* __restrict__ inv_s2p,
                                                 const float* __restrict__ dens,
                                                 const float* __restrict__ invd,
                                                 float* __restrict__ out_w,
                                                 float* __restrict__ out_d) {
  const int tile = blockIdx.x * 8 + (threadIdx.x >> 5);  // i-fast ordering
  const int i0   = (tile & (I_TILES - 1)) * 16;
  const int j0   = (tile >> 9) * 16;
  const int lane = threadIdx.x & 31;
  const int hf   = lane >> 4;
  const int rw   = lane & 15;

  const float inv_s2 = *inv_s2p;

  const float* ap = xp + (size_t)(j0 + rw) * X_DIM + 2 * hf;
  const float* bp = x  + (size_t)(i0 + rw) * X_DIM + 2 * hf;

  v8f c = {};
#pragma unroll
  for (int kk = 0; kk < 16; ++kk) {
    v2f a = *(const v2f*)(ap + 4 * kk);
    v2f b = *(const v2f*)(bp + 4 * kk);
    c = __builtin_amdgcn_wmma_f32_16x16x4_f32(false, a, false, b,
                                              (short)0, c, false, false);
  }

  const float x2v = x2[i0 + rw];
#pragma unroll
  for (int r = 0; r < 8; ++r) {
    const int j = j0 + r + 8 * hf;
    float d2 = fmaxf(xp2[j] + x2v - 2.0f * c[r], 0.0f);
    float e  = __expf(-d2 * inv_s2);
    out_w[(size_t)j * N_X + i0 + rw] = e * invd[j];
  }

  if (i0 == 0 && rw == 0) {  // emit density output once per j row
#pragma unroll
    for (int r = 0; r < 8; ++r) {
      const int j = j0 + r + 8 * hf;
      out_d[j] = dens[j];
    }
  }
}

// -------------------------------------------------------------------------
extern "C" void kernel_launch(void* const* d_in, const int* in_sizes, int n_in,
                              void* d_out, int out_size, void* d_ws, size_t ws_size,
                              hipStream_t stream) {
  const float* x   = (const float*)d_in[0];  // (8192, 64) f32
  const float* xp  = (const float*)d_in[1];  // (4096, 64) f32
  const float* sls = (const float*)d_in[2];  // scalar f32

  float* ws     = (float*)d_ws;
  float* x2     = ws;                              // 8192
  float* xp2    = ws + N_X;                        // 4096
  float* part   = xp2 + N_XP;                      // SEGS*4096
  float* dens   = part + (size_t)SEGS * N_XP;      // 4096
  float* invd   = dens + N_XP;                     // 4096
  float* inv_s2 = invd + N_XP;                     // 1

  float* out_w = (float*)d_out;                    // (4096, 8192, 1)
  float* out_d = out_w + (size_t)N_XP * N_X;       // (4096, 1)

  k_pre<<<(N_X + 255) / 256, 256, 0, stream>>>(x, xp, sls, x2, xp2, inv_s2);
  k_density<<<dim3(J_TILES, SEGS), 256, 0, stream>>>(x, xp, x2, xp2, inv_s2, part);
  k_reduce<<<(N_XP + 255) / 256, 256, 0, stream>>>(part, dens, invd);
  k_weights<<<(J_TILES * I_TILES) / 8, 256, 0, stream>>>(x, xp, x2, xp2, inv_s2,
                                                         dens, invd, out_w, out_d);
}